// EdgeSampler_62947040690666
// MI455X (gfx1250) — compile-verified
//
#include <hip/hip_runtime.h>

// EdgeSampler: one-hop neighbor sampling with replacement + valid mask.
//
// Inputs (setup_inputs order):
//   d_in[0] source_node_ids : int32 [256*512]        = 131072
//   d_in[1] row_ptr         : int32 [1000001]
//   d_in[2] col_indices     : int32 [~32M]
//   d_in[3] rand_u          : float32 [256*512*16]   = 2097152
// Output (float*, concatenated flat in return order):
//   out[0  ..  N)  src   (seed id broadcast; exact in fp32 since < 2^24)
//   out[N  .. 2N)  tgt   (sampled neighbor id)
//   out[2N .. 3N)  valid (1.0 / 0.0)
// with N = 2097152.
//
// MI455X strategy:
//  * col_indices (~128MB) + row_ptr (4MB) fit in the 192MB L2 -> gather with
//    default temporal hints so the random-access working set stays resident.
//  * rand_u (read-once) and the 24MB output are streamed non-temporally
//    (TH=NT) so they do not evict the gather working set from L2.
//  * global_prefetch_b8 on the neighbor-list head: once `start` is known the
//    (<=64-int) adjacency list is guaranteed to be gathered -> warm it while
//    rand_u loads and the index math are in flight.
//  * 4 samples per lane -> b128 coalesced rand_u loads and output stores;
//    4 lanes of each wave32 share a seed so row_ptr loads coalesce/broadcast.

#define GS_BATCH        256
#define GS_SEEDS        512
#define GS_SAMPLE       16
#define GS_N            (GS_BATCH * GS_SEEDS * GS_SAMPLE)   // 2097152 samples
#define GS_VPT          4                                   // samples per thread
#define GS_THREADS      (GS_N / GS_VPT)                     // 524288
#define GS_BLOCK        256

// Clang-native vector type: required by __builtin_nontemporal_{load,store}
// (HIP's float4 is a struct and is rejected by the builtin).
typedef float v4f __attribute__((ext_vector_type(4)));

__global__ __launch_bounds__(GS_BLOCK) void edge_sampler_kernel(
    const int*   __restrict__ seeds,     // [B*R]
    const int*   __restrict__ row_ptr,   // [NUM_NODES+1]
    const int*   __restrict__ col,       // [num_edges]
    const float* __restrict__ rand_u,    // [B*R*S]
    float*       __restrict__ out)       // [3*N]
{
    const int tid  = blockIdx.x * GS_BLOCK + threadIdx.x;
    const int base = tid * GS_VPT;                 // flat sample index (mult of 4)
    if (base >= GS_N) return;                      // grid is exact; kept for safety

    // 16 samples per seed; this thread's 4 samples all share one seed.
    const int seed_idx = base >> 4;                // / GS_SAMPLE
    const int seed     = seeds[seed_idx];

    // row_ptr[seed], row_ptr[seed+1]: 4 lanes per wave share these addresses.
    const int start = row_ptr[seed];
    const int deg   = row_ptr[seed + 1] - start;

    // gfx1250: global_prefetch_b8 — warm the neighbor list (<= 2 cachelines)
    // while the rand_u load and index arithmetic are outstanding.
    __builtin_prefetch((const void*)(col + start), 0, 3);

    // Streaming read-once data: non-temporal b128 load (TH=NT).
    const v4f u = __builtin_nontemporal_load(
        reinterpret_cast<const v4f*>(rand_u + base));

    const bool  has   = (deg > 0);
    const float degf  = (float)deg;
    const int   cap   = has ? (deg - 1) : 0;       // max(deg-1, 0)

    // idx = min((int)(u*deg), cap) — truncation toward zero matches jnp cast.
    int i0 = (int)(u.x * degf); i0 = (i0 < cap) ? i0 : cap;
    int i1 = (int)(u.y * degf); i1 = (i1 < cap) ? i1 : cap;
    int i2 = (int)(u.z * degf); i2 = (i2 < cap) ? i2 : cap;
    int i3 = (int)(u.w * degf); i3 = (i3 < cap) ? i3 : cap;

    // pos = valid ? start + idx : 0 (reference gathers col[0] for invalid).
    const int p0 = has ? (start + i0) : 0;
    const int p1 = has ? (start + i1) : 0;
    const int p2 = has ? (start + i2) : 0;
    const int p3 = has ? (start + i3) : 0;

    // L2-resident random gathers (default temporal hint).
    const int t0 = col[p0];
    const int t1 = col[p1];
    const int t2 = col[p2];
    const int t3 = col[p3];

    const float sf = (float)seed;                  // exact: seed < 2^24
    const float vf = has ? 1.0f : 0.0f;

    // Streaming write-once outputs: non-temporal b128 stores, fully coalesced.
    v4f s4 = { sf, sf, sf, sf };
    v4f t4 = { (float)t0, (float)t1, (float)t2, (float)t3 };
    v4f v4 = { vf, vf, vf, vf };

    __builtin_nontemporal_store(s4, reinterpret_cast<v4f*>(out + base));
    __builtin_nontemporal_store(t4, reinterpret_cast<v4f*>(out + GS_N + base));
    __builtin_nontemporal_store(v4, reinterpret_cast<v4f*>(out + 2 * GS_N + base));
}

extern "C" void kernel_launch(void* const* d_in, const int* in_sizes, int n_in,
                              void* d_out, int out_size, void* d_ws, size_t ws_size,
                              hipStream_t stream) {
    (void)in_sizes; (void)n_in; (void)out_size; (void)d_ws; (void)ws_size;

    const int*   seeds   = (const int*)  d_in[0];
    const int*   row_ptr = (const int*)  d_in[1];
    const int*   col     = (const int*)  d_in[2];
    const float* rand_u  = (const float*)d_in[3];
    float*       out     = (float*)      d_out;

    const int blocks = GS_THREADS / GS_BLOCK;      // 2048 blocks of 8 wave32s
    edge_sampler_kernel<<<blocks, GS_BLOCK, 0, stream>>>(
        seeds, row_ptr, col, rand_u, out);
}